// TensorCSPNetLocalClosedFormResidual_57174604645001
// MI455X (gfx1250) — compile-verified
//
#include <hip/hip_runtime.h>
#include <hip/hip_bf16.h>

// -----------------------------------------------------------------------------
// TensorCSPNet local closed-form residual, Woodbury-reduced.
//
//   logit[b,c] = (1/λ) * ( d·u  -  (L^-1 d)·(L^-1 (G u)) ),   M = G_perm + λ diag(1/w) = L Lᵀ
//   out        = base + α·β·logit
//
// d = dots of h with the 16 prototypes (permuted per class), w from group
// softmax of (2d - ||p||²), u = ±w, G = proto·protoᵀ (16x16).
//
// GEMMs (h·protoᵀ and proto·protoᵀ) use V_WMMA_F32_16X16X4_F32 chains.
// -----------------------------------------------------------------------------

typedef __attribute__((ext_vector_type(2))) float v2f;
typedef __attribute__((ext_vector_type(8))) float v8f;

#define Dk      128
#define NB      512
#define NC      4
#define NP      4
#define RIDGE   0.01f
#define INV_RIDGE 100.0f
#define TEMP_INV 1.0f
#define ALPHA   1.0f

// support index i (0..15) -> global prototype index for class c
__device__ __forceinline__ int pidx(int i, int c) {
    if (i < 4) return c * 4 + i;
    int o = i - 4;
    int blk = o >> 2;
    int cc = blk + (blk >= c ? 1 : 0);   // ascending classes != c
    return cc * 4 + (o & 3);
}

__global__ __launch_bounds__(32)
void cspnet_local_kernel(const float* __restrict__ h,
                         const float* __restrict__ proto,   // [16][128] flattened
                         const float* __restrict__ base,    // [512][4]
                         const float* __restrict__ beta,    // [1]
                         float* __restrict__ out)           // [512][4]
{
    __shared__ float dotsS[16][17];     // [row-in-tile][proto]
    __shared__ float Gs[16][17];        // proto gram
    __shared__ float triS[136][32];     // packed 16x16 lower triangle, lane-major (bank = lane)
    __shared__ float aS[16][32];        // d  -> L^-1 d
    __shared__ float bS[16][32];        // scratch l/e, then q -> L^-1 q
    __shared__ float uS[16][32];        // u = w * t

    const int lane = threadIdx.x;       // 0..31, one full wave (EXEC all ones for WMMA)
    const int lm   = lane & 15;
    const int half = lane >> 4;
    const int row0 = blockIdx.x * 16;

    // per-lane fragment pointers:  A[m][k], B[k][n]=proto[n][k], k = 4*kk + 2*half + {0,1}
    const float* hRow = h     + (size_t)(row0 + lm) * Dk + 2 * half;
    const float* pRow = proto + (size_t)lm          * Dk + 2 * half;

    v8f accD = {0.f,0.f,0.f,0.f,0.f,0.f,0.f,0.f};
    v8f accG = {0.f,0.f,0.f,0.f,0.f,0.f,0.f,0.f};

#pragma unroll
    for (int kk = 0; kk < 32; ++kk) {
        v2f av = *(const v2f*)(hRow + 4 * kk);
        v2f bv = *(const v2f*)(pRow + 4 * kk);
        // dots tile: H(16x128) x protoT(128x16)
        accD = __builtin_amdgcn_wmma_f32_16x16x4_f32(false, av, false, bv,
                                                     (short)0, accD, false, false);
        // prototype gram: proto(16x128) x protoT(128x16) (same fragment both sides)
        accG = __builtin_amdgcn_wmma_f32_16x16x4_f32(false, bv, false, bv,
                                                     (short)0, accG, false, false);
    }

    // C/D layout: vreg j, lane -> (M = j + 8*half, N = lane&15)
#pragma unroll
    for (int j = 0; j < 8; ++j) {
        dotsS[j + 8 * half][lm] = accD[j];
        Gs[j + 8 * half][lm]    = accG[j];
    }
    __syncthreads();

    const float bta = beta[0];

    // 64 (row, class) tasks per block; 2 per lane, sequential (LDS column reused)
    for (int t = 0; t < 2; ++t) {
        const int c = half + 2 * t;
        const int m = lm;

        // ---- softmax logits l = (2d - ||p||^2)/TEMP, per-group max ----
        float mx0 = -1e30f, mx1 = -1e30f;
        for (int i = 0; i < 16; ++i) {
            int   pi = pidx(i, c);
            float dd = dotsS[m][pi];
            float li = (2.0f * dd - Gs[pi][pi]) * TEMP_INV;
            aS[i][lane] = dd;
            bS[i][lane] = li;
            if (i < 4) mx0 = fmaxf(mx0, li); else mx1 = fmaxf(mx1, li);
        }
        float s0 = 0.f, s1 = 0.f;
        for (int i = 0; i < 16; ++i) {
            float e = __expf(bS[i][lane] - (i < 4 ? mx0 : mx1));
            uS[i][lane] = e;
            if (i < 4) s0 += e; else s1 += e;
        }
        const float r0 = 0.5f / s0, r1 = 0.5f / s1;   // SAME_MASS / OPP_MASS

        // ---- u = w * t (t = +1 same, -1 opp);  sdu = d·u ----
        float sdu = 0.f;
        for (int i = 0; i < 16; ++i) {
            float w  = uS[i][lane] * (i < 4 ? r0 : r1);
            float ui = (i < 4) ? w : -w;
            uS[i][lane] = ui;
            sdu += aS[i][lane] * ui;
        }

        // ---- q = G_perm u  (into bS);  M = G_perm + λ diag(1/w) (into triS) ----
        for (int i = 0; i < 16; ++i) {
            int pi = pidx(i, c);
            float qi = 0.f;
            for (int j = 0; j < 16; ++j)
                qi += Gs[pi][pidx(j, c)] * uS[j][lane];
            bS[i][lane] = qi;
            for (int j = 0; j < i; ++j)
                triS[i * (i + 1) / 2 + j][lane] = Gs[pi][pidx(j, c)];
            float wi = fabsf(uS[i][lane]);           // |u| == w > 0
            triS[i * (i + 1) / 2 + i][lane] = Gs[pi][pi] + RIDGE / wi;
        }

        // ---- right-looking Cholesky with fused forward substitutions ----
        // after loop: aS = L^-1 d, bS = L^-1 q
        for (int j = 0; j < 16; ++j) {
            float ljj = sqrtf(triS[j * (j + 1) / 2 + j][lane]);
            float inv = 1.0f / ljj;
            float aj = aS[j][lane] * inv; aS[j][lane] = aj;
            float bj = bS[j][lane] * inv; bS[j][lane] = bj;
            for (int i = j + 1; i < 16; ++i) {
                float lij = triS[i * (i + 1) / 2 + j][lane] * inv;
                triS[i * (i + 1) / 2 + j][lane] = lij;
                aS[i][lane] -= lij * aj;
                bS[i][lane] -= lij * bj;
                for (int k2 = j + 1; k2 <= i; ++k2)
                    triS[i * (i + 1) / 2 + k2][lane] -=
                        lij * triS[k2 * (k2 + 1) / 2 + j][lane];
            }
        }

        float sab = 0.f;
        for (int i = 0; i < 16; ++i)
            sab += aS[i][lane] * bS[i][lane];

        float logit = (sdu - sab) * INV_RIDGE;
        int   oi    = (row0 + m) * NC + c;
        out[oi] = base[oi] + ALPHA * bta * logit;
    }
}

extern "C" void kernel_launch(void* const* d_in, const int* in_sizes, int n_in,
                              void* d_out, int out_size, void* d_ws, size_t ws_size,
                              hipStream_t stream) {
    const float* h     = (const float*)d_in[0];   // [512,128]
    const float* proto = (const float*)d_in[1];   // [4,4,128]
    const float* base  = (const float*)d_in[2];   // [512,4]
    const float* beta  = (const float*)d_in[3];   // [1]
    float* out = (float*)d_out;                   // [512,4]

    dim3 grid(NB / 16);   // 32 workgroups
    dim3 block(32);       // one wave32 each
    hipLaunchKernelGGL(cspnet_local_kernel, grid, block, 0, stream,
                       h, proto, base, beta, out);
}